// CatAttention_16329465660071
// MI455X (gfx1250) — compile-verified
//
#include <hip/hip_runtime.h>
#include <math.h>

#define BATCH  4
#define NCTX   1024
#define DIN    512
#define NHEADS 8
#define DHEAD  64
#define NVARS  8
#define SPITCH 1025   // 16-row score strip pitch (bank-conflict padding)

typedef float v2f __attribute__((ext_vector_type(2)));
typedef float v8f __attribute__((ext_vector_type(8)));

// D = A(16x4,f32) * B(4x16,f32) + C(16x16,f32), wave32.
__device__ inline v8f wmma_f32(v2f a, v2f b, v8f c) {
  return __builtin_amdgcn_wmma_f32_16x16x4_f32(
      /*neg_a=*/false, a, /*neg_b=*/false, b,
      /*c_mod=*/(short)0, c, /*reuse_a=*/false, /*reuse_b=*/false);
}

// ---------------------------------------------------------------------------
// Kernel A: softmaxes of the tiny weight tensors.
//   pK/pQ/pV[i][v] = softmax_v(W[i,:])      (8x8 each)
//   Wp[i][e][f]    = softmax_f(W_pred[i,e,:]) (8x64x64)
// ---------------------------------------------------------------------------
__global__ void prep_kernel(const float* __restrict__ WK, const float* __restrict__ WQ,
                            const float* __restrict__ WV, const float* __restrict__ Wpred,
                            float* __restrict__ pK, float* __restrict__ pQ,
                            float* __restrict__ pV, float* __restrict__ Wp) {
  int t = threadIdx.x;
  if (t < 24) {
    const float* src = (t < 8) ? WK : (t < 16 ? WQ : WV);
    float*       dst = (t < 8) ? pK : (t < 16 ? pQ : pV);
    int row = t & 7;
    float mx = -1e30f;
    for (int v = 0; v < 8; ++v) mx = fmaxf(mx, src[row * 8 + v]);
    float e[8]; float s = 0.f;
    for (int v = 0; v < 8; ++v) { e[v] = expf(src[row * 8 + v] - mx); s += e[v]; }
    float inv = 1.f / s;
    for (int v = 0; v < 8; ++v) dst[row * 8 + v] = e[v] * inv;
  }
  for (int row = t; row < NHEADS * DHEAD; row += blockDim.x) {
    const float* src = Wpred + row * 64;
    float*       dst = Wp + row * 64;
    float mx = -1e30f;
    for (int f = 0; f < 64; ++f) mx = fmaxf(mx, src[f]);
    float s = 0.f;
    for (int f = 0; f < 64; ++f) s += expf(src[f] - mx);
    float inv = 1.f / s;
    for (int f = 0; f < 64; ++f) dst[f] = expf(src[f] - mx) * inv;
  }
}

// ---------------------------------------------------------------------------
// Kernel B1: per-head variable mixtures.
//   K (stored TRANSPOSED: KT[b,i,h,p]), V[b,i,p,h], XQ[b,i,p,h]
//   One thread per (b,i,p,h); reuses the 8 strided x-loads for all three.
// ---------------------------------------------------------------------------
__global__ void proj_mix_kernel(const float* __restrict__ x,
                                const float* __restrict__ pK, const float* __restrict__ pQ,
                                const float* __restrict__ pV,
                                float* __restrict__ KT, float* __restrict__ Vt,
                                float* __restrict__ XQ) {
  int idx = blockIdx.x * blockDim.x + threadIdx.x;   // [0, B*H*S*D)
  int h = idx & 63;
  int p = (idx >> 6) & 1023;
  int i = (idx >> 16) & 7;
  int b = idx >> 19;
  const float* xp = x + ((size_t)b * NCTX + p) * DIN + h;
  float xv[8];
#pragma unroll
  for (int v = 0; v < 8; ++v) xv[v] = xp[v * 64];
  float k = 0.f, vv = 0.f, xq = 0.f;
#pragma unroll
  for (int v = 0; v < 8; ++v) {
    k  += pK[i * 8 + v] * xv[v];
    vv += pV[i * 8 + v] * xv[v];
    xq += pQ[i * 8 + v] * xv[v];
  }
  size_t bi = (size_t)b * NHEADS + i;
  KT[(bi * 64 + h) * NCTX + p]  = k;
  Vt[(bi * NCTX + p) * 64 + h]  = vv;
  XQ[(bi * NCTX + p) * 64 + h]  = xq;
}

// ---------------------------------------------------------------------------
// Kernel B2: Q[b,i,p,f] = sum_c XQ[b,i,p,c] * Wp[i,c,f]  via f32 WMMA.
//   One wave per (b,i,16-row tile). 64 WMMA ops.
// ---------------------------------------------------------------------------
__global__ void qproj_kernel(const float* __restrict__ XQ, const float* __restrict__ Wp,
                             float* __restrict__ Qt) {
  int bid = blockIdx.x;            // B*H*(S/16)
  int pt = bid & 63;
  int i  = (bid >> 6) & 7;
  int b  = bid >> 9;
  int lane = threadIdx.x;
  int m = lane & 15, hi = lane >> 4;
  size_t bi = (size_t)b * NHEADS + i;

  const float* arow = XQ + (bi * NCTX + pt * 16 + m) * 64;
  v2f a[16];
#pragma unroll
  for (int kk = 0; kk < 16; ++kk) a[kk] = *(const v2f*)(arow + 4 * kk + 2 * hi);

  v8f c[4] = {};
  const float* wpb = Wp + i * 4096;
#pragma unroll
  for (int kk = 0; kk < 16; ++kk) {
    int k0 = 4 * kk + 2 * hi;
#pragma unroll
    for (int t = 0; t < 4; ++t) {
      v2f bb;
      bb.x = wpb[(k0 + 0) * 64 + 16 * t + m];
      bb.y = wpb[(k0 + 1) * 64 + 16 * t + m];
      c[t] = wmma_f32(a[kk], bb, c[t]);
    }
  }
  float* qrow = Qt + (bi * NCTX + pt * 16) * 64;
#pragma unroll
  for (int t = 0; t < 4; ++t)
#pragma unroll
    for (int r = 0; r < 8; ++r)
      qrow[(r + 8 * hi) * 64 + 16 * t + m] = c[t][r];
}

// ---------------------------------------------------------------------------
// Kernel C: causal attention. One wave per (b, head, 16-query tile).
//   Pass 1: pre = Q_tile * K^T (WMMA) -> LDS strip; accumulate srow (masked).
//   Pass 2: read strip in WMMA-A layout, w = exp((log(pre'+1e-20)+bias)/8),
//           denom += w, Z += w * V (WMMA). pre' adds relu(1-srow/(srow+1e-10))
//           to key column 0. bias(d=q-p): d==0 -> -2, else -(d-1)/1023.
// ---------------------------------------------------------------------------
__global__ void attn_kernel(const float* __restrict__ Qt, const float* __restrict__ KT,
                            const float* __restrict__ Vt, float* __restrict__ out) {
  extern __shared__ float smem[];
  float* pre_s  = smem;                    // 16 * SPITCH
  float* srow_s = smem + 16 * SPITCH;      // 16
  float* den_s  = srow_s + 16;             // 16

  int bid = blockIdx.x;            // B*H*(S/16)
  int qt = bid & 63;
  int i  = (bid >> 6) & 7;
  int b  = bid >> 9;
  int q0 = qt * 16;
  int lane = threadIdx.x;
  int m = lane & 15, hi = lane >> 4;
  size_t bi = (size_t)b * NHEADS + i;

  // Q tile in WMMA-A layout: lane (m,hi), reg pair j -> Q[q0+m][4*kk+2*hi+j]
  const float* qrow = Qt + (bi * NCTX + q0 + m) * 64;
  v2f qa[16];
#pragma unroll
  for (int kk = 0; kk < 16; ++kk) qa[kk] = *(const v2f*)(qrow + 4 * kk + 2 * hi);

  const float* ktb = KT + bi * 64 * NCTX;
  const float* vtb = Vt + bi * (size_t)NCTX * 64;
  int nch = qt + 1;                // causal: keys [0, q0+16)

  // ---- Pass 1: scores into LDS + masked row sums -------------------------
  float srow[8];
#pragma unroll
  for (int r = 0; r < 8; ++r) srow[r] = 0.f;

  for (int ch = 0; ch < nch; ++ch) {
    int pbase = ch * 16;
    v8f acc = {};
#pragma unroll
    for (int kk = 0; kk < 16; ++kk) {
      int k0 = 4 * kk + 2 * hi;
      v2f kb;                              // B[k'][n] = KT[h=k'][p=pbase+n]
      kb.x = ktb[(size_t)(k0 + 0) * NCTX + pbase + m];
      kb.y = ktb[(size_t)(k0 + 1) * NCTX + pbase + m];
      acc = wmma_f32(qa[kk], kb, acc);
    }
    int p = pbase + m;                     // C/D layout: row r+8*hi, col m
#pragma unroll
    for (int r = 0; r < 8; ++r) {
      int row = r + 8 * hi;
      pre_s[row * SPITCH + p] = acc[r];
      if (p <= q0 + row) srow[r] += acc[r];
    }
  }
  // reduce srow over the 16 key-columns held by lanes of the same half
#pragma unroll
  for (int r = 0; r < 8; ++r) {
    float s = srow[r];
    s += __shfl_xor(s, 1, 32);
    s += __shfl_xor(s, 2, 32);
    s += __shfl_xor(s, 4, 32);
    s += __shfl_xor(s, 8, 32);
    srow[r] = s;
  }
  if (m == 0) {
#pragma unroll
    for (int r = 0; r < 8; ++r) srow_s[8 * hi + r] = srow[r];
  }
  __syncthreads();

  // ---- Pass 2: weights + Z = w * V ---------------------------------------
  float srm  = srow_s[m];
  float adj0 = fmaxf(1.0f - srm / (srm + 1e-10f), 0.0f);
  int q = q0 + m;

  float denom = 0.f;
  v8f zc[4] = {};

  for (int ch = 0; ch < nch; ++ch) {
    int pbase = ch * 16;
    v2f wa[4];
#pragma unroll
    for (int kk = 0; kk < 4; ++kk) {
#pragma unroll
      for (int j = 0; j < 2; ++j) {
        int kc = 4 * kk + 2 * hi + j;      // A layout: A[m][kc]
        int p  = pbase + kc;
        float w = 0.f;
        if (p <= q) {
          float pr = pre_s[m * SPITCH + p];
          if (p == 0) pr += adj0;
          float bias = (p == q) ? -2.0f : -(float)(q - p - 1) * (1.0f / 1023.0f);
          w = expf((logf(pr + 1e-20f) + bias) * 0.125f);
        }
        denom += w;
        wa[kk][j] = w;
      }
    }
#pragma unroll
    for (int t = 0; t < 4; ++t) {
#pragma unroll
      for (int kk = 0; kk < 4; ++kk) {
        int k0 = 4 * kk + 2 * hi;
        v2f vb;                            // B[k'][n] = V[p=pbase+k'][h=16t+n]
        vb.x = vtb[(size_t)(pbase + k0 + 0) * 64 + 16 * t + m];
        vb.y = vtb[(size_t)(pbase + k0 + 1) * 64 + 16 * t + m];
        zc[t] = wmma_f32(wa[kk], vb, zc[t]);
      }
    }
  }
  denom += __shfl_xor(denom, 16, 32);      // halves cover disjoint col subsets
  if (hi == 0) den_s[m] = denom;
  __syncthreads();

  // out[b, q, i*64 + h], h = 16*t + m ; Z rows in C/D layout: r + 8*hi
  float* orow = out + ((size_t)b * NCTX + q0) * DIN + i * 64;
#pragma unroll
  for (int t = 0; t < 4; ++t)
#pragma unroll
    for (int r = 0; r < 8; ++r) {
      int row = r + 8 * hi;
      orow[(size_t)row * DIN + 16 * t + m] = zc[t][r] / den_s[row];
    }
}

// ---------------------------------------------------------------------------
extern "C" void kernel_launch(void* const* d_in, const int* in_sizes, int n_in,
                              void* d_out, int out_size, void* d_ws, size_t ws_size,
                              hipStream_t stream) {
  const float* x      = (const float*)d_in[0];
  // d_in[1] = mask (bool, causal tril) -- structure known, not needed
  const float* W_K_W  = (const float*)d_in[2];
  const float* W_Q_W  = (const float*)d_in[3];
  const float* W_V_W  = (const float*)d_in[4];
  const float* W_pred = (const float*)d_in[5];
  float* out = (float*)d_out;

  float* ws = (float*)d_ws;
  float* pK = ws;                 // 64
  float* pQ = pK + 64;            // 64
  float* pV = pQ + 64;            // 64
  float* Wp = pV + 64;            // 8*64*64 = 32768
  float* KT = Wp + 32768;         // B*H*64*NCTX = 2097152
  float* Vt = KT + 2097152;       // 2097152
  float* XQ = Vt + 2097152;       // 2097152
  float* Qt = XQ + 2097152;       // 2097152   (total ~33.7 MB)

  prep_kernel<<<1, 256, 0, stream>>>(W_K_W, W_Q_W, W_V_W, W_pred, pK, pQ, pV, Wp);
  proj_mix_kernel<<<(BATCH * NHEADS * NCTX * DHEAD) / 256, 256, 0, stream>>>(
      x, pK, pQ, pV, KT, Vt, XQ);
  qproj_kernel<<<BATCH * NHEADS * (NCTX / 16), 32, 0, stream>>>(XQ, Wp, Qt);
  size_t lds_bytes = (size_t)(16 * SPITCH + 32) * sizeof(float);
  attn_kernel<<<BATCH * NHEADS * (NCTX / 16), 32, lds_bytes, stream>>>(Qt, KT, Vt, out);
}